// MeanPool_7327214207175
// MI455X (gfx1250) — compile-verified
//
#include <hip/hip_runtime.h>

typedef __attribute__((ext_vector_type(2))) float v2f;
typedef __attribute__((ext_vector_type(8))) float v8f;

#define NUM_SEQS        16
#define SEQ_LEN         8192
#define HIDDEN          1024
#define ROWS_PER_BLOCK  256
#define BLOCKS_PER_SEG  (SEQ_LEN / ROWS_PER_BLOCK)   // 32
#define THREADS         256                          // 8 wave32s
#define WAVES           (THREADS / 32)
#define COLS_PER_WAVE   (HIDDEN / WAVES)             // 128
#define COLGROUPS       (COLS_PER_WAVE / 32)         // 4 groups of two 16-col tiles

__global__ void meanpool_zero_kernel(float* __restrict__ out) {
    int i = blockIdx.x * blockDim.x + threadIdx.x;
    if (i < NUM_SEQS * HIDDEN) out[i] = 0.0f;
}

__global__ __launch_bounds__(THREADS)
void meanpool_wmma_kernel(const float* __restrict__ hs,
                          const long long* __restrict__ lens,
                          float* __restrict__ out) {
    const int seg   = blockIdx.x / BLOCKS_PER_SEG;
    const int chunk = blockIdx.x % BLOCKS_PER_SEG;
    const int wave  = threadIdx.x >> 5;
    const int lane  = threadIdx.x & 31;
    const int g     = lane >> 4;     // half-wave group (0/1)
    const int n     = lane & 15;     // column within 16-wide tile

    const float inv = 1.0f / (float)lens[seg];
    const size_t row0 = (size_t)seg * SEQ_LEN + (size_t)chunk * ROWS_PER_BLOCK;

    // A = 16x4 all-ones: column sums appear in every row of D.
    const v2f ones = {1.0f, 1.0f};

    for (int cg = 0; cg < COLGROUPS; ++cg) {
        const int col0 = wave * COLS_PER_WAVE + cg * 32;
        v8f c0 = {};  // accumulator, cols [col0, col0+16)
        v8f c1 = {};  // accumulator, cols [col0+16, col0+32)

        // Each 16-lane group supplies 2 of the 4 K-rows per WMMA step.
        const float* p = hs + row0 * HIDDEN + (size_t)col0 + n
                            + (size_t)g * 2 * HIDDEN;

        #pragma unroll 4
        for (int r = 0; r < ROWS_PER_BLOCK; r += 4) {
            const float* q = p + (size_t)r * HIDDEN;
            // 16 lanes x (two tiles) = 128B contiguous, aligned, per row; NT:
            // 512MB streamed once, don't pollute the 192MB L2.
            float b00 = __builtin_nontemporal_load(q);
            float b01 = __builtin_nontemporal_load(q + 16);
            float b10 = __builtin_nontemporal_load(q + HIDDEN);
            float b11 = __builtin_nontemporal_load(q + HIDDEN + 16);
            v2f bA = {b00, b10};
            v2f bB = {b01, b11};
            c0 = __builtin_amdgcn_wmma_f32_16x16x4_f32(
                     false, ones, false, bA, (short)0, c0, false, false);
            c1 = __builtin_amdgcn_wmma_f32_16x16x4_f32(
                     false, ones, false, bB, (short)0, c1, false, false);
        }

        // D VGPR0: lanes 0-15 hold M=0 row = column sums (all D rows equal).
        if (lane < 16) {
            float* o = out + (size_t)seg * HIDDEN + col0 + n;
            unsafeAtomicAdd(o,      c0[0] * inv);
            unsafeAtomicAdd(o + 16, c1[0] * inv);
        }
    }
}

extern "C" void kernel_launch(void* const* d_in, const int* in_sizes, int n_in,
                              void* d_out, int out_size, void* d_ws, size_t ws_size,
                              hipStream_t stream) {
    const float*     hs   = (const float*)d_in[0];
    const long long* lens = (const long long*)d_in[1];
    float*           out  = (float*)d_out;

    (void)in_sizes; (void)n_in; (void)out_size; (void)d_ws; (void)ws_size;

    meanpool_zero_kernel<<<(NUM_SEQS * HIDDEN + 255) / 256, 256, 0, stream>>>(out);
    meanpool_wmma_kernel<<<NUM_SEQS * BLOCKS_PER_SEG, THREADS, 0, stream>>>(hs, lens, out);
}